// IMLESubsetkLayer_53592601919727
// MI455X (gfx1250) — compile-verified
//
#include <hip/hip_runtime.h>
#include <cstdint>

// ---------------- problem constants (reference is fixed-shape) --------------
constexpr int BB  = 40;            // batch rows
constexpr int NN  = 32768;         // items per row
constexpr int LCH = 64;            // items per chunk
constexpr int CC  = NN / LCH;      // 512 chunks per row
constexpr int GG  = CC / 32;       // 16 groups of 32 chunks (stitch)

typedef __attribute__((ext_vector_type(2))) float v2f;
typedef __attribute__((ext_vector_type(8))) float v8f;

// ---------------- helpers ---------------------------------------------------
__device__ __forceinline__ float clipped_sigmoid(float x) {
  // p = exp(min(log_sigmoid(x), -1e-7)) == min(sigmoid(x), exp(-1e-7))
  float p = 1.0f / (1.0f + __expf(-x));
  return fminf(p, 0.99999990f);
}

__device__ __forceinline__ float wave_max32(float v) {
#pragma unroll
  for (int m = 16; m >= 1; m >>= 1) v = fmaxf(v, __shfl_xor(v, m, 32));
  return v;
}
__device__ __forceinline__ float wave_sum32(float v) {
#pragma unroll
  for (int m = 16; m >= 1; m >>= 1) v += __shfl_xor(v, m, 32);
  return v;
}

__device__ __forceinline__ float rng_uniform(uint32_t b, uint32_t i) {
  uint32_t h = b * 0x9E3779B1u + i * 0x85EBCA77u + 0x165667B1u;
  h ^= h >> 16; h *= 0x7FEB352Du; h ^= h >> 15; h *= 0x846CA68Bu; h ^= h >> 16;
  return (float)(h >> 8) * 5.9604645e-8f;   // [0,1)
}

// D = A(16x16) * B(16x16), row-major tiles in LDS, via 4 chained
// v_wmma_f32_16x16x4_f32 (FP32 WMMA — needed for probability dynamic range).
__device__ __forceinline__ v8f matmul16_lds(const float* A, const float* B) {
  const int lane = threadIdx.x & 31;
  const int half = lane >> 4;
  const int idx  = lane & 15;
  v8f c = {};
#pragma unroll
  for (int k = 0; k < 16; k += 4) {
    v2f a, b;
    // A fragment: 16x4 slice; lanes 0-15 carry K=k,k+1 ; lanes 16-31 K=k+2,k+3
    a.x = A[idx * 16 + (k + 2 * half)];
    a.y = A[idx * 16 + (k + 2 * half + 1)];
    // B fragment: 4x16 slice; rows striped across lane halves
    b.x = B[(k + 2 * half) * 16 + idx];
    b.y = B[(k + 2 * half + 1) * 16 + idx];
    c = __builtin_amdgcn_wmma_f32_16x16x4_f32(false, a, false, b,
                                              (short)0, c, false, false);
  }
  return c;
}

__device__ __forceinline__ void store_c_lds(float* T, v8f c) {
  const int lane = threadIdx.x & 31;
  const int half = lane >> 4;
  const int n    = lane & 15;
#pragma unroll
  for (int v = 0; v < 8; ++v) T[(v + 8 * half) * 16 + n] = c[v];
}

// ---------------- K1: per-chunk 16x16 transition matrices (WMMA combine) ----
__global__ __launch_bounds__(32)
void k1_chunk_mats(const float* __restrict__ logits, float* __restrict__ chunkM) {
  const int chunkId = blockIdx.x;              // b*CC + c
  const int lane = threadIdx.x;
  const int col  = lane & 15;
  __shared__ float sub[4][256];
  __shared__ float T[256];

  const long base = (long)(chunkId / CC) * NN + (long)(chunkId % CC) * LCH;
  float pa = clipped_sigmoid(logits[base + lane]);
  float pb = clipped_sigmoid(logits[base + lane + 32]);

  // 4 sub-matrices of 16 items each: short serial bidiagonal chains.
#pragma unroll 1
  for (int s = 0; s < 4; ++s) {
    float m[16];
#pragma unroll
    for (int r = 0; r < 16; ++r) m[r] = (r == col) ? 1.0f : 0.0f;
#pragma unroll 1
    for (int t = 0; t < 16; ++t) {
      int item = s * 16 + t;
      float p = (item < 32) ? __shfl(pa, item, 32) : __shfl(pb, item - 32, 32);
      float q = 1.0f - p;
#pragma unroll
      for (int r = 15; r >= 1; --r) m[r] = q * m[r] + p * m[r - 1];
      m[0] *= q;
    }
    if (lane < 16) {
#pragma unroll
      for (int r = 0; r < 16; ++r) sub[s][r * 16 + col] = m[r];
    }
  }
  __syncthreads();
  for (int e = lane; e < 256; e += 32) T[e] = sub[0][e];
  __syncthreads();
  // M = M3*M2*M1*M0 via chained 16x16x16 WMMA products (EXEC all-ones here).
#pragma unroll 1
  for (int s = 1; s < 4; ++s) {
    v8f acc = matmul16_lds(&sub[s][0], &T[0]);
    __syncthreads();
    store_c_lds(&T[0], acc);
    __syncthreads();
  }
  float* out = chunkM + (size_t)chunkId * 256;
  for (int e = lane; e < 256; e += 32) out[e] = T[e];
}

// ---------------- K2: per-row chunk-level prefix/suffix vector scans --------
__global__ __launch_bounds__(32)
void k2_scan(const float* __restrict__ chunkM,
             float* __restrict__ prefixS, float* __restrict__ suffixT) {
  const int b    = blockIdx.x;
  const int lane = threadIdx.x;
  const int j    = lane & 15;

  if (blockIdx.y == 0) {
    // forward: s_{c+1} = M_c * s_c, renormalized (ratios make scales moot)
    float s = (j == 1 && lane < 16) ? 1.0f : 0.0f;
    if (lane < 16) prefixS[((size_t)b * (CC + 1)) * 16 + j] = s;
#pragma unroll 1
    for (int c = 0; c < CC; ++c) {
      const float* M = chunkM + ((size_t)b * CC + c) * 256;
      if (c + 1 < CC) __builtin_prefetch(M + 256, 0, 1);   // global_prefetch_b8
      float acc = 0.0f;
#pragma unroll
      for (int t = 0; t < 16; ++t) acc += M[j * 16 + t] * __shfl(s, t, 32);
      float ns = (lane < 16) ? acc : 0.0f;
      float mx = wave_max32(ns);
      if (mx > 0.0f) ns /= mx;
      s = ns;
      if (lane < 16) prefixS[((size_t)b * (CC + 1) + (c + 1)) * 16 + j] = s;
    }
  } else {
    // backward adjoint: t_c = t_{c+1} * M_c (row-vector), renormalized
    float t = (j == 11 && lane < 16) ? 1.0f : 0.0f;
    if (lane < 16) suffixT[((size_t)b * (CC + 1) + CC) * 16 + j] = t;
#pragma unroll 1
    for (int c = CC - 1; c >= 0; --c) {
      const float* M = chunkM + ((size_t)b * CC + c) * 256;
      if (c > 0) __builtin_prefetch(M - 256, 0, 1);
      float acc = 0.0f;
#pragma unroll
      for (int r = 0; r < 16; ++r) acc += __shfl(t, r, 32) * M[r * 16 + j];
      float nt = (lane < 16) ? acc : 0.0f;
      float mx = wave_max32(nt);
      if (mx > 0.0f) nt /= mx;
      t = nt;
      if (lane < 16) suffixT[((size_t)b * (CC + 1) + c) * 16 + j] = t;
    }
  }
}

// ---------------- K3: per-chunk marginals y + speculative sampling ----------
__global__ __launch_bounds__(32)
void k3_chunk(const float* __restrict__ logits,
              const float* __restrict__ prefixS, const float* __restrict__ suffixT,
              float* __restrict__ yArr, uint32_t* __restrict__ spec) {
  const int chunkId = blockIdx.x;
  const int b = chunkId / CC, c = chunkId % CC;
  const int lane = threadIdx.x;

  __shared__ float f[LCH + 1][16];
  __shared__ float rho[LCH + 1];
  __shared__ float pArr[LCH];

  const long base = (long)b * NN + (long)c * LCH;
  float pa = clipped_sigmoid(logits[base + lane]);
  float pb = clipped_sigmoid(logits[base + lane + 32]);
  pArr[lane]      = pa;
  pArr[lane + 32] = pb;

  // local forward states (renormalized; rho[t] records the per-step scale)
  float cur = (lane < 16) ? prefixS[((size_t)b * (CC + 1) + c) * 16 + lane] : 0.0f;
  if (lane < 16) f[0][lane] = cur;
  if (lane == 0) rho[0] = 1.0f;
#pragma unroll 1
  for (int t = 1; t <= LCH; ++t) {
    int item = t - 1;
    float p = (item < 32) ? __shfl(pa, item, 32) : __shfl(pb, item - 32, 32);
    float q = 1.0f - p;
    float up = __shfl(cur, (lane + 31) & 31, 32);
    if (lane == 0) up = 0.0f;
    float nx = q * cur + p * up;
    if (lane >= 16) nx = 0.0f;
    float mx = wave_max32(nx);
    float sc = (mx > 0.0f) ? mx : 1.0f;
    nx /= sc;
    cur = nx;
    if (lane < 16) f[t][lane] = nx;
    if (lane == 0) rho[t] = sc;
  }
  __syncthreads();

  // adjoint sweep: y_i = p*(g·S f)/(q*(g·f) + p*(g·S f)) — scale-free ratio
  float g = (lane < 16) ? suffixT[((size_t)b * (CC + 1) + c + 1) * 16 + lane] : 0.0f;
  float y0 = 0.0f, y1 = 0.0f;
#pragma unroll 1
  for (int t = LCH; t >= 1; --t) {
    int item = t - 1;
    float p = (item < 32) ? __shfl(pa, item, 32) : __shfl(pb, item - 32, 32);
    float q = 1.0f - p;
    float fm1  = (lane < 16) ? f[t - 1][lane] : 0.0f;
    float fm1s = (lane > 0 && lane < 16) ? f[t - 1][lane - 1] : 0.0f;
    float num = wave_sum32(g * fm1s);
    float den = wave_sum32(g * fm1);
    float y = p * num / (q * den + p * num + 1e-30f);
    if (item < 32) { if (lane == item)      y0 = y; }
    else           { if (lane == item - 32) y1 = y; }
    float gp1 = __shfl(g, (lane + 1) & 31, 32);
    float ng = q * g + p * gp1;
    if (lane >= 16) ng = 0.0f;
    float mg = wave_max32(ng);
    if (mg > 0.0f) ng /= mg;
    g = ng;
  }
  yArr[base + lane]      = y0;
  yArr[base + lane + 32] = y1;
  __syncthreads();

  // speculative backward sampling for every possible entry state j0 in [1,11]
  if (lane >= 1 && lane <= 11) {
    int j = lane;
    uint32_t bits0 = 0, bits1 = 0;
#pragma unroll 1
    for (int t = LCH; t >= 1; --t) {
      int item = t - 1;
      float p    = pArr[item];
      float fnum = f[t - 1][j - 1];
      float fden = f[t][j] * rho[t];
      float pr   = (fden > 0.0f) ? (p * fnum / fden) : 0.0f;
      float u    = rng_uniform((uint32_t)b, (uint32_t)(c * LCH + item));
      if (u < pr) {
        if (item < 32) bits0 |= (1u << item); else bits1 |= (1u << (item - 32));
        j = (j > 1) ? j - 1 : 1;
      }
    }
    uint32_t* sp = spec + ((size_t)chunkId * 12 + lane) * 3;
    sp[0] = (uint32_t)j; sp[1] = bits0; sp[2] = bits1;
  }
}

// ---------------- K4: stitch speculative chunk maps (2-level) ---------------
__global__ __launch_bounds__(256)
void k4_stitch(const uint32_t* __restrict__ spec, uint32_t* __restrict__ realized) {
  const int b = blockIdx.x;
  const int t = threadIdx.x;
  __shared__ uint8_t H[GG][16];
  __shared__ uint8_t S[GG];
  int g = t >> 4, s = t & 15;
  uint32_t x = (uint32_t)s;
  if (s >= 1 && s <= 11) {
    for (int i = 31; i >= 0; --i)
      x = spec[(((size_t)b * CC + (g * 32 + i)) * 12 + x) * 3];
  }
  H[g][s] = (uint8_t)x;
  __syncthreads();
  if (t == 0) {
    uint32_t xx = 11;  // sampler starts at column k+1
    for (int g2 = GG - 1; g2 >= 0; --g2) { S[g2] = (uint8_t)xx; xx = H[g2][xx]; }
  }
  __syncthreads();
  if (t < GG) {
    uint32_t xx = S[t];
    for (int i = 31; i >= 0; --i) {
      int c = t * 32 + i;
      realized[(size_t)b * CC + c] = xx;
      xx = spec[(((size_t)b * CC + c) * 12 + xx) * 3];
    }
  }
}

// ---------------- K5: emit (sample - y) + y ---------------------------------
__global__ __launch_bounds__(256)
void k5_out(const float* __restrict__ yArr, const uint32_t* __restrict__ spec,
            const uint32_t* __restrict__ realized, float* __restrict__ out) {
  size_t idx = (size_t)blockIdx.x * blockDim.x + threadIdx.x;
  if (idx >= (size_t)BB * NN) return;
  int b = (int)(idx / NN), i = (int)(idx % NN);
  int c = i >> 6, t = i & 63;
  uint32_t j0 = realized[(size_t)b * CC + c];
  const uint32_t* sp = spec + (((size_t)b * CC + c) * 12 + j0) * 3;
  uint32_t bit = (t < 32) ? ((sp[1] >> t) & 1u) : ((sp[2] >> (t - 32)) & 1u);
  float y = yArr[idx];
  out[idx] = ((float)bit - y) + y;
}

// ---------------- launcher --------------------------------------------------
extern "C" void kernel_launch(void* const* d_in, const int* in_sizes, int n_in,
                              void* d_out, int out_size, void* d_ws, size_t ws_size,
                              hipStream_t stream) {
  (void)in_sizes; (void)n_in; (void)out_size; (void)ws_size;
  const float* logits = (const float*)d_in[0];
  float* out = (float*)d_out;

  char* ws = (char*)d_ws;
  float*    chunkM   = (float*)ws;                               // BB*CC*256
  float*    prefixS  = chunkM  + (size_t)BB * CC * 256;          // BB*(CC+1)*16
  float*    suffixT  = prefixS + (size_t)BB * (CC + 1) * 16;     // BB*(CC+1)*16
  float*    yArr     = suffixT + (size_t)BB * (CC + 1) * 16;     // BB*NN
  uint32_t* spec     = (uint32_t*)(yArr + (size_t)BB * NN);      // BB*CC*12*3
  uint32_t* realized = spec + (size_t)BB * CC * 12 * 3;          // BB*CC

  k1_chunk_mats<<<BB * CC, 32, 0, stream>>>(logits, chunkM);
  k2_scan<<<dim3(BB, 2), 32, 0, stream>>>(chunkM, prefixS, suffixT);
  k3_chunk<<<BB * CC, 32, 0, stream>>>(logits, prefixS, suffixT, yArr, spec);
  k4_stitch<<<BB, 256, 0, stream>>>(spec, realized);
  k5_out<<<(BB * NN + 255) / 256, 256, 0, stream>>>(yArr, spec, realized, out);
}